// SituationGCN_3770981286014
// MI455X (gfx1250) — compile-verified
//
#include <hip/hip_runtime.h>
#include <hip/hip_bf16.h>

typedef __attribute__((ext_vector_type(2))) float v2f;
typedef __attribute__((ext_vector_type(8))) float v8f;

#define HID    128
#define NCLS   11
#define INCH   15
#define NGRAPH 64

__device__ __forceinline__ void atomAddF(float* p, float v) {
    __hip_atomic_fetch_add(p, v, __ATOMIC_RELAXED, __HIP_MEMORY_SCOPE_AGENT);
}

// ---------------- init: zero agg buffer, deg=1 (self loop), zero pooled/counts
__global__ void k_init(float* __restrict__ B, float* __restrict__ deg,
                       float* __restrict__ pooled, float* __restrict__ counts, int N) {
    const unsigned i = blockIdx.x * 256u + threadIdx.x;
    const unsigned tot = (unsigned)N * HID;
    if (i < tot)          B[i] = 0.0f;
    if (i < (unsigned)N)  deg[i] = 1.0f;
    if (i < NGRAPH * HID) pooled[i] = 0.0f;
    if (i < NGRAPH)       counts[i] = 0.0f;
}

__global__ void k_zero(float* __restrict__ p, int n) {
    const int i = blockIdx.x * 256 + threadIdx.x;
    if (i < n) p[i] = 0.0f;
}

// ---------------- degree count on dst, then deg -> deg^{-1/2}
__global__ void k_degree(const int* __restrict__ dst, float* __restrict__ deg, int E) {
    const int e = blockIdx.x * 256 + threadIdx.x;
    if (e < E) atomAddF(&deg[dst[e]], 1.0f);
}

__global__ void k_dinv(float* __restrict__ deg, int N) {
    const int i = blockIdx.x * 256 + threadIdx.x;
    if (i < N) deg[i] = 1.0f / __builtin_sqrtf(deg[i]);
}

// ---------------- layer-1 transform: out[N,128] = x[N,15] @ W1[15,128] (fp32 WMMA, K padded to 16)
__global__ void k_gemm1(const float* __restrict__ x, const float* __restrict__ W1,
                        float* __restrict__ out, int N) {
    const int lane = threadIdx.x & 31;
    const int wave = threadIdx.x >> 5;
    const int row0 = blockIdx.x * 16;
    const int col0 = wave * 16;
    const int half = lane >> 4;     // 0: K pair {k,k+1}, 1: K pair {k+2,k+3}
    const int l15  = lane & 15;
    const int rowL = min(row0 + l15, N - 1);   // clamp loads; N%16==0 so stores are in-range

    v8f acc = {};
#pragma unroll
    for (int k0 = 0; k0 < 16; k0 += 4) {
        const int ka = k0 + 2 * half;
        v2f a, b;
        a.x = (ka     < INCH) ? x[rowL * INCH + ka]     : 0.0f;
        a.y = (ka + 1 < INCH) ? x[rowL * INCH + ka + 1] : 0.0f;
        b.x = (ka     < INCH) ? W1[ka * HID + col0 + l15]       : 0.0f;
        b.y = (ka + 1 < INCH) ? W1[(ka + 1) * HID + col0 + l15] : 0.0f;
        acc = __builtin_amdgcn_wmma_f32_16x16x4_f32(false, a, false, b, (short)0, acc, false, false);
    }
#pragma unroll
    for (int r = 0; r < 8; ++r) {
        const int row = row0 + r + 8 * half;
        if (row < N) out[(size_t)row * HID + col0 + l15] = acc[r];
    }
}

// ---------------- layer-2 transform: out[N,128] = h[N,128] @ W2[128,128], h tile staged in LDS
__global__ void k_gemm2(const float* __restrict__ h, const float* __restrict__ W2,
                        float* __restrict__ out, int N) {
    __shared__ __align__(16) float tile[16][132];   // 132 pad -> bank = 4*row + k (conflict-free)
    const int row0 = blockIdx.x * 16;

    // cooperative load of 16x128 floats (512 float4 / 256 threads)
#pragma unroll
    for (int j = 0; j < 2; ++j) {
        const int f4 = threadIdx.x + j * 256;       // float4 index 0..511
        const int r  = f4 >> 5;                     // 32 float4 per row
        const int c4 = f4 & 31;
        const int row = min(row0 + r, N - 1);
        const float4 v = ((const float4*)(h + (size_t)row * HID))[c4];
        *(float4*)&tile[r][c4 * 4] = v;
    }
    __syncthreads();

    const int lane = threadIdx.x & 31;
    const int wave = threadIdx.x >> 5;
    const int col0 = wave * 16;
    const int half = lane >> 4;
    const int l15  = lane & 15;

    v8f acc = {};
#pragma unroll 4
    for (int k0 = 0; k0 < HID; k0 += 4) {
        const int ka = k0 + 2 * half;
        v2f a, b;
        a.x = tile[l15][ka];
        a.y = tile[l15][ka + 1];
        b.x = W2[ka * HID + col0 + l15];
        b.y = W2[(ka + 1) * HID + col0 + l15];
        acc = __builtin_amdgcn_wmma_f32_16x16x4_f32(false, a, false, b, (short)0, acc, false, false);
    }
#pragma unroll
    for (int r = 0; r < 8; ++r) {
        const int row = row0 + r + 8 * half;
        if (row < N) out[(size_t)row * HID + col0 + l15] = acc[r];
    }
}

// ---------------- edge scatter: agg[dst] += dinv[src]*dinv[dst] * m[src]  (wave per edge)
__global__ void k_scatter(const float* __restrict__ m, const int* __restrict__ src,
                          const int* __restrict__ dst, const float* __restrict__ dinv,
                          float* __restrict__ agg, int E) {
    const int lane = threadIdx.x & 31;
    const int e = blockIdx.x * 8 + (threadIdx.x >> 5);
    if (e >= E) return;
    const int s = src[e], d = dst[e];
    const float nrm = dinv[s] * dinv[d];
    const float4 v = ((const float4*)(m + (size_t)s * HID))[lane];
    float* ad = agg + (size_t)d * HID + lane * 4;
    atomAddF(ad + 0, v.x * nrm);
    atomAddF(ad + 1, v.y * nrm);
    atomAddF(ad + 2, v.z * nrm);
    atomAddF(ad + 3, v.w * nrm);
}

// ---------------- add self-loop term + bias, ReLU, in place on agg (wave per node)
__global__ void k_relu_self(const float* __restrict__ m, float* __restrict__ agg,
                            const float* __restrict__ dinv, const float* __restrict__ bias, int N) {
    const int lane = threadIdx.x & 31;
    const int node = blockIdx.x * 8 + (threadIdx.x >> 5);
    if (node >= N) return;
    const float di = dinv[node];
    const float sl = di * di;
    const float4 mv = ((const float4*)(m + (size_t)node * HID))[lane];
    float4* ap = (float4*)(agg + (size_t)node * HID) + lane;
    float4 av = *ap;
    const float4 bv = ((const float4*)bias)[lane];
    av.x = fmaxf(av.x + sl * mv.x + bv.x, 0.0f);
    av.y = fmaxf(av.y + sl * mv.y + bv.y, 0.0f);
    av.z = fmaxf(av.z + sl * mv.z + bv.z, 0.0f);
    av.w = fmaxf(av.w + sl * mv.w + bv.w, 0.0f);
    *ap = av;
}

// ---------------- layer-2 ReLU fused with mean-pool accumulation
__global__ void k_relu_pool(const float* __restrict__ m, const float* __restrict__ agg,
                            const float* __restrict__ dinv, const float* __restrict__ bias,
                            const int* __restrict__ batch,
                            float* __restrict__ pooled, float* __restrict__ counts, int N) {
    const int lane = threadIdx.x & 31;
    const int node = blockIdx.x * 8 + (threadIdx.x >> 5);
    if (node >= N) return;
    const float di = dinv[node];
    const float sl = di * di;
    const float4 mv = ((const float4*)(m + (size_t)node * HID))[lane];
    const float4 av = ((const float4*)(agg + (size_t)node * HID))[lane];
    const float4 bv = ((const float4*)bias)[lane];
    const float hx = fmaxf(av.x + sl * mv.x + bv.x, 0.0f);
    const float hy = fmaxf(av.y + sl * mv.y + bv.y, 0.0f);
    const float hz = fmaxf(av.z + sl * mv.z + bv.z, 0.0f);
    const float hw = fmaxf(av.w + sl * mv.w + bv.w, 0.0f);
    const int g = batch[node];
    float* pp = pooled + g * HID + lane * 4;
    atomAddF(pp + 0, hx);
    atomAddF(pp + 1, hy);
    atomAddF(pp + 2, hz);
    atomAddF(pp + 3, hw);
    if (lane == 0) atomAddF(&counts[g], 1.0f);
}

// ---------------- out[G,11] = (pooled/cnt) @ Wf + bf
__global__ void k_final(const float* __restrict__ pooled, const float* __restrict__ counts,
                        const float* __restrict__ Wf, const float* __restrict__ bf,
                        float* __restrict__ out) {
    const int t = blockIdx.x * 256 + threadIdx.x;
    if (t >= NGRAPH * NCLS) return;
    const int g = t / NCLS, c = t % NCLS;
    const float inv = 1.0f / fmaxf(counts[g], 1.0f);
    float s = 0.0f;
#pragma unroll 8
    for (int k = 0; k < HID; ++k) s += pooled[g * HID + k] * Wf[k * NCLS + c];
    out[t] = s * inv + bf[c];
}

extern "C" void kernel_launch(void* const* d_in, const int* in_sizes, int n_in,
                              void* d_out, int out_size, void* d_ws, size_t ws_size,
                              hipStream_t stream) {
    const float* x   = (const float*)d_in[0];
    const int*   ei  = (const int*)d_in[1];
    const int*   bat = (const int*)d_in[2];
    const float* W1  = (const float*)d_in[3];
    const float* b1  = (const float*)d_in[4];
    const float* W2  = (const float*)d_in[5];
    const float* b2  = (const float*)d_in[6];
    const float* Wf  = (const float*)d_in[7];
    const float* bfv = (const float*)d_in[8];
    float* out = (float*)d_out;

    const int N = in_sizes[2];
    const int E = in_sizes[1] / 2;
    const int* src = ei;
    const int* dst = ei + E;

    float* A      = (float*)d_ws;                 // m buffer      N*128
    float* B      = A + (size_t)N * HID;          // agg/h buffer  N*128
    float* dinv   = B + (size_t)N * HID;          // deg -> dinv   N
    float* pooled = dinv + N;                     // 64*128
    float* counts = pooled + NGRAPH * HID;        // 64

    const int tot = N * HID;
    k_init   <<<(tot + 255) / 256, 256, 0, stream>>>(B, dinv, pooled, counts, N);
    k_degree <<<(E + 255) / 256,   256, 0, stream>>>(dst, dinv, E);
    k_dinv   <<<(N + 255) / 256,   256, 0, stream>>>(dinv, N);

    // layer 1
    k_gemm1    <<<(N + 15) / 16, 256, 0, stream>>>(x, W1, A, N);
    k_scatter  <<<(E + 7) / 8,   256, 0, stream>>>(A, src, dst, dinv, B, E);
    k_relu_self<<<(N + 7) / 8,   256, 0, stream>>>(A, B, dinv, b1, N);

    // layer 2
    k_gemm2    <<<(N + 15) / 16, 256, 0, stream>>>(B, W2, A, N);
    k_zero     <<<(tot + 255) / 256, 256, 0, stream>>>(B, tot);
    k_scatter  <<<(E + 7) / 8,   256, 0, stream>>>(A, src, dst, dinv, B, E);
    k_relu_pool<<<(N + 7) / 8,   256, 0, stream>>>(A, B, dinv, b2, bat, pooled, counts, N);

    // readout
    k_final<<<(NGRAPH * NCLS + 255) / 256, 256, 0, stream>>>(pooled, counts, Wf, bfv, out);
}